// HiGATELoss_77738908057987
// MI455X (gfx1250) — compile-verified
//
#include <hip/hip_runtime.h>
#include <hip/hip_bf16.h>

typedef __attribute__((ext_vector_type(2))) float v2f;
typedef __attribute__((ext_vector_type(4))) float v4f;
typedef __attribute__((ext_vector_type(8))) float v8f;

#define SPATIAL_WEIGHT 0.01f
#define NUM_CLASSES 10
#define KCL 32

// ---------------- Focal loss: one thread per sample ----------------
__global__ void focal_kernel(const float* __restrict__ logits,
                             const int* __restrict__ targets,
                             const int* __restrict__ tissue,
                             const float* __restrict__ cw,
                             const float* __restrict__ tw,
                             float* __restrict__ acc, int B) {
    int b = blockIdx.x * blockDim.x + threadIdx.x;
    float v = 0.0f;
    if (b < B) {
        const float* l = logits + (size_t)b * NUM_CLASSES;
        float m = l[0];
        #pragma unroll
        for (int c = 1; c < NUM_CLASSES; ++c) m = fmaxf(m, l[c]);
        float s = 0.0f;
        #pragma unroll
        for (int c = 0; c < NUM_CLASSES; ++c) s += __expf(l[c] - m);
        float lse = m + __logf(s);
        int   t   = targets[b];
        float lpt = l[t] - lse;           // log p_t
        float pt  = __expf(lpt);
        float om  = 1.0f - pt;
        float fw  = om * om * cw[t] * tw[tissue[b]];   // GAMMA == 2
        v = -fw * lpt;
    }
    // wave32 reduction
    #pragma unroll
    for (int off = 16; off > 0; off >>= 1) v += __shfl_xor(v, off, 32);
    if ((threadIdx.x & 31) == 0) atomicAdd(acc, v);
}

// ------------- Edge term: 16 edges per wave-step via WMMA -------------
// pair(i,j,k) = (Si>0 && Sj>0) ? Si*Sj : 0  ==  relu(Si)*relu(Sj)
// C = A x B with A row m = relu(S[src_m]) chunk, B col n = relu(S[dst_n]) chunk;
// diagonal C[i][i] = dot(relu(S[src_i]), relu(S[dst_i])).
__global__ void edge_wmma_kernel(const float* __restrict__ S,
                                 const float* __restrict__ pos,
                                 const int* __restrict__ ei,
                                 float* __restrict__ acc, int E) {
    const int lane   = threadIdx.x & 31;
    const int r16    = lane & 15;          // edge-in-chunk this lane serves
    const int hi     = lane >> 4;          // which half of K this lane supplies
    const int waveId = blockIdx.x * (blockDim.x >> 5) + (threadIdx.x >> 5);
    const int nWaves = gridDim.x * (blockDim.x >> 5);
    const int nChunk = E >> 4;

    float facc = 0.0f;

    for (int c = waveId; c < nChunk; c += nWaves) {
        const int e   = (c << 4) + r16;
        const int src = ei[e];
        const int dst = ei[E + e];

        // 16 contiguous K values per node per lane: exactly-once gather of S.
        const v4f* pa = (const v4f*)(S + (size_t)src * KCL + hi * 16);
        const v4f* pb = (const v4f*)(S + (size_t)dst * KCL + hi * 16);
        float sa[16], sb[16];
        #pragma unroll
        for (int i = 0; i < 4; ++i) {
            v4f va = pa[i], vb = pb[i];
            sa[4*i+0] = fmaxf(va.x, 0.0f); sb[4*i+0] = fmaxf(vb.x, 0.0f);
            sa[4*i+1] = fmaxf(va.y, 0.0f); sb[4*i+1] = fmaxf(vb.y, 0.0f);
            sa[4*i+2] = fmaxf(va.z, 0.0f); sb[4*i+2] = fmaxf(vb.z, 0.0f);
            sa[4*i+3] = fmaxf(va.w, 0.0f); sb[4*i+3] = fmaxf(vb.w, 0.0f);
        }

        // squared distance for edge (c*16 + r16) — every lane computes its own
        v2f ps = *(const v2f*)(pos + 2 * (size_t)src);
        v2f pd = *(const v2f*)(pos + 2 * (size_t)dst);
        float dx = ps.x - pd.x, dy = ps.y - pd.y;
        float d2 = dx * dx + dy * dy;

        // 8 chained f32 WMMAs cover K=32 (K-label permutation is legal since
        // A and B use identical lane->k mapping; dot is order-invariant).
        v8f cm = {};
        #pragma unroll
        for (int w = 0; w < 8; ++w) {
            v2f av; av.x = sa[2*w]; av.y = sa[2*w + 1];
            v2f bv; bv.x = sb[2*w]; bv.y = sb[2*w + 1];
            cm = __builtin_amdgcn_wmma_f32_16x16x4_f32(
                    false, av, false, bv, (short)0, cm, false, false);
        }

        // Diagonal lives on lanes 0..7 (M=N=d in VGPR d) and 24..31 (M=N=8+d).
        // Those lanes' own d2 (edge lane&15) is exactly the matching edge.
        float cv = cm[lane & 7];
        bool onDiag = (lane < 8) || (lane >= 24);
        facc += onDiag ? cv * d2 : 0.0f;
    }

    // scalar tail (E % 16), no WMMA here
    const int tail = E & 15;
    if (tail && waveId == 0 && lane < tail) {
        const int e = (nChunk << 4) + lane;
        const int src = ei[e], dst = ei[E + e];
        float s = 0.0f;
        #pragma unroll
        for (int k = 0; k < KCL; ++k)
            s += fmaxf(S[(size_t)src * KCL + k], 0.0f) *
                 fmaxf(S[(size_t)dst * KCL + k], 0.0f);
        float dx = pos[2*(size_t)src]     - pos[2*(size_t)dst];
        float dy = pos[2*(size_t)src + 1] - pos[2*(size_t)dst + 1];
        facc += s * (dx * dx + dy * dy);
    }

    #pragma unroll
    for (int off = 16; off > 0; off >>= 1) facc += __shfl_xor(facc, off, 32);
    if (lane == 0) atomicAdd(acc, facc);
}

// ---------------- Finalize: (total, cls, spatial) ----------------
__global__ void finalize_kernel(const float* __restrict__ ws,
                                float* __restrict__ out, int B, int E) {
    if (threadIdx.x == 0 && blockIdx.x == 0) {
        float cls = ws[0] / (float)B;
        float sp  = ws[1] * SPATIAL_WEIGHT / (float)E;
        out[0] = cls + sp;
        out[1] = cls;
        out[2] = sp;
    }
}

extern "C" void kernel_launch(void* const* d_in, const int* in_sizes, int n_in,
                              void* d_out, int out_size, void* d_ws, size_t ws_size,
                              hipStream_t stream) {
    const float* logits  = (const float*)d_in[0];
    const int*   targets = (const int*)  d_in[1];
    const float* S       = (const float*)d_in[2];
    const float* pos     = (const float*)d_in[3];
    const int*   ei      = (const int*)  d_in[4];
    const int*   tissue  = (const int*)  d_in[5];
    const float* cw      = (const float*)d_in[6];
    const float* tw      = (const float*)d_in[7];
    float*       out     = (float*)d_out;
    float*       ws      = (float*)d_ws;

    const int B = in_sizes[1];
    const int E = in_sizes[4] / 2;

    hipMemsetAsync(ws, 0, 2 * sizeof(float), stream);

    focal_kernel<<<(B + 255) / 256, 256, 0, stream>>>(
        logits, targets, tissue, cw, tw, ws + 0, B);

    // persistent waves; memory-bound loop, ~12 chunks/wave at 1024 blocks
    edge_wmma_kernel<<<1024, 256, 0, stream>>>(S, pos, ei, ws + 1, E);

    finalize_kernel<<<1, 32, 0, stream>>>(ws, out, B, E);
}